// GraphSN_84911503442024
// MI455X (gfx1250) — compile-verified
//
#include <hip/hip_runtime.h>
#include <math.h>

typedef __attribute__((ext_vector_type(2))) float v2f;
typedef __attribute__((ext_vector_type(8))) float v8f;

constexpr int NB = 8;
constexpr int NN = 2048;
constexpr int DD = 128;
constexpr int HH = 128;
constexpr float NEG_BIG = -9000000000000000.0f;
constexpr float BN_EPS  = 1e-5f;

#if defined(__has_builtin)
#  if __has_builtin(__builtin_amdgcn_global_load_async_to_lds_b128)
#    define USE_ASYNC_LDS 1
#  endif
#endif
#ifndef USE_ASYNC_LDS
#  define USE_ASYNC_LDS 0
#endif

#if USE_ASYNC_LDS
typedef int ivec4 __attribute__((vector_size(16)));
typedef __attribute__((address_space(1))) ivec4* gptr_iv4;
typedef __attribute__((address_space(3))) ivec4* lptr_iv4;
__device__ __forceinline__ void wait_async0() {
#  if __has_builtin(__builtin_amdgcn_s_wait_asynccnt)
  __builtin_amdgcn_s_wait_asynccnt(0);
#  else
  asm volatile("s_wait_asynccnt 0" ::: "memory");
#  endif
}
#endif

__device__ __forceinline__ v8f wmma4(v2f a, v2f b, v8f c) {
  // D = A(16x4 f32) * B(4x16 f32) + C(16x16 f32)
  return __builtin_amdgcn_wmma_f32_16x16x4_f32(false, a, false, b, (short)0, c,
                                               false, false);
}

// ---------------------------------------------------------------------------
// Dense GEMM:  Out(16384x128) = Xin(16384x128) @ Wm(128x128)  [+bias][relu]
// One wave per 16-row slab, 8 column tiles, K loop in steps of 4 via WMMA f32.
// ---------------------------------------------------------------------------
template <int BIAS, int RELU>
__global__ void gemm_dense(const float* __restrict__ Xin,
                           const float* __restrict__ Wm,
                           const float* __restrict__ bias,
                           float* __restrict__ Out) {
  const int lane = threadIdx.x & 31;
  const int half = lane >> 4;
  const int lr   = lane & 15;
  const int koff = half * 2;
  const int mBase = blockIdx.x * 16;
  const int r = mBase + lr;

  v8f acc[8] = {};
  for (int k0 = 0; k0 < DD; k0 += 4) {
    const int kk = k0 + koff;
    v2f a = *(const v2f*)(Xin + (size_t)r * DD + kk);
#pragma unroll
    for (int nt = 0; nt < 8; ++nt) {
      const int n = nt * 16 + lr;
      v2f b;
      b.x = Wm[kk * DD + n];
      b.y = Wm[(kk + 1) * DD + n];
      acc[nt] = wmma4(a, b, acc[nt]);
    }
  }
#pragma unroll
  for (int nt = 0; nt < 8; ++nt) {
    const int n = nt * 16 + lr;
#pragma unroll
    for (int rr = 0; rr < 8; ++rr) {
      const int row = mBase + rr + 8 * half;
      float v = acc[nt][rr];
      if (BIAS) v += bias[n];
      if (RELU) v = v > 0.0f ? v : 0.0f;
      Out[(size_t)row * DD + n] = v;
    }
  }
}

// ---------------------------------------------------------------------------
// A-side GEMM per batch:  Out(2048x128) = Alike(2048x2048) @ Bmat(2048x128)
// MODE 0: Alike = A2  (A with diagonal scaled by eps)             -> new_feat
// MODE 1: Alike = softmax probabilities rebuilt on the fly        -> h_prime
// 4 waves / block, B chunks (64x128 = 32KB) staged through LDS via the
// CDNA5 async global->LDS path (ASYNCcnt) when available.
// ---------------------------------------------------------------------------
template <int MODE>
__global__ void gemm_attA(const float* __restrict__ A,
                          const float* __restrict__ Bmat,
                          const float* __restrict__ epsP,
                          const float* __restrict__ s1,
                          const float* __restrict__ s2,
                          const float* __restrict__ mrow,
                          const float* __restrict__ lrow,
                          float* __restrict__ Out) {
  __shared__ float lds[64 * DD];
  const int tid  = threadIdx.x;
  const int lane = tid & 31;
  const int wave = tid >> 5;
  const int batch = blockIdx.x >> 5;   // 32 groups per batch
  const int grp   = blockIdx.x & 31;
  const int mBase = (grp * 4 + wave) * 16;
  const int half = lane >> 4;
  const int lr   = lane & 15;
  const int koff = half * 2;
  const int r = mBase + lr;

  const float* Arow = A + (size_t)batch * NN * NN + (size_t)r * NN;
  const float* Bb   = Bmat + (size_t)batch * NN * DD;
  const float* s2b  = s2 ? (s2 + (size_t)batch * NN) : nullptr;
  const float epsv  = epsP[0];
  float s1v = 0.0f, mv = 0.0f, invl = 0.0f;
  if (MODE == 1) {
    s1v  = s1[(size_t)batch * NN + r];
    mv   = mrow[(size_t)batch * NN + r];
    invl = 1.0f / lrow[(size_t)batch * NN + r];
  }

  v8f acc[8] = {};
  for (int c = 0; c < NN / 64; ++c) {
    const int kbase = c * 64;
    {  // cooperative contiguous copy of the B chunk into LDS
      const float4* src = (const float4*)(Bb + (size_t)kbase * DD);
      float4* dst = (float4*)lds;
#if USE_ASYNC_LDS
      for (int i = tid; i < 64 * DD / 4; i += 128) {
        __builtin_amdgcn_global_load_async_to_lds_b128(
            (gptr_iv4)(src + i), (lptr_iv4)(dst + i),
            /*offset=*/0, /*cpol=*/0);
      }
      wait_async0();
#else
      for (int i = tid; i < 64 * DD / 4; i += 128) dst[i] = src[i];
#endif
    }
    __syncthreads();
    if (c + 1 < NN / 64)  // pull next A chunk toward L2
      __builtin_prefetch(Arow + kbase + 64 + lane * 2, 0, 1);

    for (int ks = 0; ks < 16; ++ks) {
      const int kk = ks * 4 + koff;
      const int j  = kbase + kk;
      v2f av = *(const v2f*)(Arow + j);
      v2f a;
      const float a2x = (j == r)     ? epsv * av.x : av.x;
      const float a2y = (j + 1 == r) ? epsv * av.y : av.y;
      if (MODE == 0) {
        a.x = a2x;
        a.y = a2y;
      } else {
        v2f s2p = *(const v2f*)(s2b + j);
        float ex, ey;
        if (a2x != 0.0f) { float e = s1v + s2p.x; ex = e > 0.0f ? e : 0.2f * e; }
        else ex = NEG_BIG;
        if (a2y != 0.0f) { float e = s1v + s2p.y; ey = e > 0.0f ? e : 0.2f * e; }
        else ey = NEG_BIG;
        a.x = __expf(ex - mv) * invl;
        a.y = __expf(ey - mv) * invl;
      }
#pragma unroll
      for (int nt = 0; nt < 8; ++nt) {
        const int n = nt * 16 + lr;
        v2f b;
        b.x = lds[kk * DD + n];
        b.y = lds[(kk + 1) * DD + n];
        acc[nt] = wmma4(a, b, acc[nt]);
      }
    }
    __syncthreads();
  }

  float* Ob = Out + (size_t)batch * NN * DD;
#pragma unroll
  for (int nt = 0; nt < 8; ++nt) {
    const int n = nt * 16 + lr;
#pragma unroll
    for (int rr = 0; rr < 8; ++rr) {
      const int row = mBase + rr + 8 * half;
      Ob[(size_t)row * DD + n] = acc[nt][rr];
    }
  }
}

// ---------------------------------------------------------------------------
// s1 = new_feat @ a1 ; s2 = new_feat @ a2   (one 128-thread block per row)
// ---------------------------------------------------------------------------
__global__ void score_kernel(const float* __restrict__ nf,
                             const float* __restrict__ avec,
                             float* __restrict__ s1, float* __restrict__ s2) {
  __shared__ float sh1[128], sh2[128];
  const int row = blockIdx.x;
  const int t = threadIdx.x;
  const float v = nf[(size_t)row * DD + t];
  sh1[t] = v * avec[t];
  sh2[t] = v * avec[DD + t];
  __syncthreads();
  for (int s = 64; s > 0; s >>= 1) {
    if (t < s) { sh1[t] += sh1[t + s]; sh2[t] += sh2[t + s]; }
    __syncthreads();
  }
  if (t == 0) { s1[row] = sh1[0]; s2[row] = sh2[0]; }
}

// ---------------------------------------------------------------------------
// Per-row softmax stats: m = rowmax(masked e), l = sum exp(e - m)
// ---------------------------------------------------------------------------
__global__ void softstats_kernel(const float* __restrict__ A,
                                 const float* __restrict__ epsP,
                                 const float* __restrict__ s1,
                                 const float* __restrict__ s2,
                                 float* __restrict__ mrow,
                                 float* __restrict__ lrow) {
  __shared__ float red[256];
  const int t = threadIdx.x;
  const int rowG = blockIdx.x;
  const int batch = rowG >> 11;
  const int i = rowG & (NN - 1);
  const float* Arow = A + (size_t)batch * NN * NN + (size_t)i * NN;
  const float* s2b = s2 + (size_t)batch * NN;
  const float epsv = epsP[0];
  const float s1v = s1[rowG];

  float e[8];
  float mx = NEG_BIG;
#pragma unroll
  for (int q = 0; q < 8; ++q) {
    const int j = t + q * 256;
    const float av = Arow[j];
    const float a2 = (j == i) ? epsv * av : av;
    float ev;
    if (a2 != 0.0f) { float x = s1v + s2b[j]; ev = x > 0.0f ? x : 0.2f * x; }
    else ev = NEG_BIG;
    e[q] = ev;
    mx = fmaxf(mx, ev);
  }
  red[t] = mx;
  __syncthreads();
  for (int s = 128; s > 0; s >>= 1) {
    if (t < s) red[t] = fmaxf(red[t], red[t + s]);
    __syncthreads();
  }
  const float m = red[0];
  __syncthreads();
  float sum = 0.0f;
#pragma unroll
  for (int q = 0; q < 8; ++q) sum += __expf(e[q] - m);
  red[t] = sum;
  __syncthreads();
  for (int s = 128; s > 0; s >>= 1) {
    if (t < s) red[t] += red[t + s];
    __syncthreads();
  }
  if (t == 0) { mrow[rowG] = m; lrow[rowG] = red[0]; }
}

// ---------------------------------------------------------------------------
// BatchNorm over axes (B, H) for each n; in place. One 256-thread block per n.
// Two-pass variance for numerical agreement with jnp.var.
// ---------------------------------------------------------------------------
__global__ void bn_kernel(float* __restrict__ X, const float* __restrict__ gamma,
                          const float* __restrict__ beta) {
  __shared__ float red[256];
  const int t = threadIdx.x;
  const int n = blockIdx.x;
  float v[4];
  float sum = 0.0f;
#pragma unroll
  for (int q = 0; q < 4; ++q) {
    const int idx = t + q * 256;  // 0..1023 over (b,h)
    const int b = idx >> 7;
    const int h = idx & 127;
    const float x = X[((size_t)b * NN + n) * HH + h];
    v[q] = x; sum += x;
  }
  red[t] = sum;
  __syncthreads();
  for (int s = 128; s > 0; s >>= 1) {
    if (t < s) red[t] += red[t + s];
    __syncthreads();
  }
  const float mean = red[0] * (1.0f / 1024.0f);
  __syncthreads();
  float dev = 0.0f;
#pragma unroll
  for (int q = 0; q < 4; ++q) { const float d = v[q] - mean; dev += d * d; }
  red[t] = dev;
  __syncthreads();
  for (int s = 128; s > 0; s >>= 1) {
    if (t < s) red[t] += red[t + s];
    __syncthreads();
  }
  const float var = red[0] * (1.0f / 1024.0f);
  const float sc = gamma[n] * rsqrtf(var + BN_EPS);
  const float bt = beta[n];
#pragma unroll
  for (int q = 0; q < 4; ++q) {
    const int idx = t + q * 256;
    const int b = idx >> 7;
    const int h = idx & 127;
    X[((size_t)b * NN + n) * HH + h] = sc * (v[q] - mean) + bt;
  }
}

// ---------------------------------------------------------------------------
extern "C" void kernel_launch(void* const* d_in, const int* in_sizes, int n_in,
                              void* d_out, int out_size, void* d_ws, size_t ws_size,
                              hipStream_t stream) {
  (void)in_sizes; (void)n_in; (void)out_size; (void)ws_size;
  const float* A   = (const float*)d_in[0];
  const float* X   = (const float*)d_in[1];
  const float* W   = (const float*)d_in[2];
  const float* av  = (const float*)d_in[3];
  const float* eps = (const float*)d_in[4];
  const float* W1  = (const float*)d_in[5];
  const float* b1  = (const float*)d_in[6];
  const float* g1  = (const float*)d_in[7];
  const float* be1 = (const float*)d_in[8];
  const float* W2  = (const float*)d_in[9];
  const float* b2  = (const float*)d_in[10];
  const float* g2  = (const float*)d_in[11];
  const float* be2 = (const float*)d_in[12];
  const float* Wl  = (const float*)d_in[13];
  const float* bl  = (const float*)d_in[14];
  float* out = (float*)d_out;
  float* ws  = (float*)d_ws;

  const size_t FMAT = (size_t)NB * NN * DD;  // 2,097,152 floats
  float* xw = ws;                 // X@W, later reused for h1
  float* nf = ws + FMAT;          // new_feat, later reused for h2
  float* hp = ws + 2 * FMAT;      // h_prime
  float* s1 = ws + 3 * FMAT;
  float* s2 = s1 + (size_t)NB * NN;
  float* mr = s2 + (size_t)NB * NN;
  float* lr = mr + (size_t)NB * NN;

  const int rowBlocks = NB * NN / 16;  // 1024

  // 1. XW = X @ W
  gemm_dense<0, 0><<<dim3(rowBlocks), dim3(32), 0, stream>>>(X, W, nullptr, xw);
  // 2. new_feat = A2 @ XW
  gemm_attA<0><<<dim3(NB * 32), dim3(128), 0, stream>>>(A, xw, eps, nullptr,
                                                        nullptr, nullptr, nullptr, nf);
  // 3. attention scores s1, s2
  score_kernel<<<dim3(NB * NN), dim3(128), 0, stream>>>(nf, av, s1, s2);
  // 4. softmax row stats (max, sum)
  softstats_kernel<<<dim3(NB * NN), dim3(256), 0, stream>>>(A, eps, s1, s2, mr, lr);
  // 5. h_prime = softmax(att) @ new_feat  (P rebuilt on the fly)
  gemm_attA<1><<<dim3(NB * 32), dim3(128), 0, stream>>>(A, nf, eps, s1, s2, mr, lr, hp);
  // 6. h1 = relu(h_prime @ W1 + b1)
  gemm_dense<1, 1><<<dim3(rowBlocks), dim3(32), 0, stream>>>(hp, W1, b1, xw);
  // 7. BN1 (per-n over batch & feature)
  bn_kernel<<<dim3(NN), dim3(256), 0, stream>>>(xw, g1, be1);
  // 8. h2 = relu(h1 @ W2 + b2)
  gemm_dense<1, 1><<<dim3(rowBlocks), dim3(32), 0, stream>>>(xw, W2, b2, nf);
  // 9. BN2
  bn_kernel<<<dim3(NN), dim3(256), 0, stream>>>(nf, g2, be2);
  // 10. out = relu(h2 @ Wl + bl)
  gemm_dense<1, 1><<<dim3(rowBlocks), dim3(32), 0, stream>>>(nf, Wl, bl, out);
}